// _ParametricLayer_13975823581507
// MI455X (gfx1250) — compile-verified
//
#include <hip/hip_runtime.h>

typedef __attribute__((ext_vector_type(2))) float v2f;
typedef __attribute__((ext_vector_type(4))) float v4f;
typedef __attribute__((ext_vector_type(8))) float v8f;

#define NN    4096
#define FIN   256
#define HEADS 8
#define SLOPE 0.2f

__device__ __forceinline__ float lrelu(float v) { return v > 0.0f ? v : SLOPE * v; }

// ---------------------------------------------------------------------------
// Kernel 1: fold a into w.  WA[k][n], k=0..255 (=f), n=0..15.
//   n<8 : WA[f][n]   = sum_g w[f,g,n]   * a[g]
//   n>=8: WA[f][n]   = sum_g w[f,g,n-8] * a[64+g]
// w layout (F_IN, F, H) row-major: w[f*512 + g*8 + h]
// ---------------------------------------------------------------------------
__global__ void gat_wa(const float* __restrict__ w, const float* __restrict__ a,
                       float* __restrict__ wa) {
  const int f = threadIdx.x;  // 256 threads, 1 block
  const float* wf = w + f * (64 * HEADS);
  float s1[HEADS], s2[HEADS];
#pragma unroll
  for (int h = 0; h < HEADS; ++h) { s1[h] = 0.0f; s2[h] = 0.0f; }
  for (int gg = 0; gg < 64; ++gg) {
    const float a1 = a[gg];
    const float a2 = a[64 + gg];
#pragma unroll
    for (int h = 0; h < HEADS; ++h) {
      const float wv = wf[gg * HEADS + h];
      s1[h] += wv * a1;
      s2[h] += wv * a2;
    }
  }
#pragma unroll
  for (int h = 0; h < HEADS; ++h) {
    wa[f * 16 + h]     = s1[h];
    wa[f * 16 + 8 + h] = s2[h];
  }
}

// ---------------------------------------------------------------------------
// Kernel 2: E = X (4096x256) * WA (256x16) via V_WMMA_F32_16X16X4_F32.
// One wave32 per 16-row tile, 64 WMMA steps over K=256.
// A frag (16x4): lane<16 holds K={k,k+1}, lane>=16 holds K={k+2,k+3}, M=lane&15
// B frag (4x16): v0 = row (k + 2*(lane>>4)), v1 = next row, N = lane&15
// C/D (16x16):  vgpr r -> M = r + 8*(lane>>4), N = lane&15
// Outputs: EI[i*8 + h] (n<8), EJT[h*4096 + j] (n>=8, transposed for streaming)
// ---------------------------------------------------------------------------
__global__ void gat_proj(const float* __restrict__ x, const float* __restrict__ wa,
                         float* __restrict__ ei, float* __restrict__ ejt) {
  const int lane  = threadIdx.x;        // 0..31 (wave32, blockDim=32)
  const int m0    = blockIdx.x * 16;    // 256 tiles
  const int mrow  = m0 + (lane & 15);
  const int khalf = (lane >> 4) * 2;    // 0 or 2
  const int ncol  = lane & 15;

  v8f acc = {};
  const float* ap = x + (size_t)mrow * FIN + khalf;
  const float* bp = wa + khalf * 16 + ncol;
#pragma unroll 4
  for (int k = 0; k < FIN; k += 4) {
    v2f av; av.x = ap[0]; av.y = ap[1];
    v2f bv; bv.x = bp[0]; bv.y = bp[16];
    acc = __builtin_amdgcn_wmma_f32_16x16x4_f32(
        /*neg_a=*/false, av, /*neg_b=*/false, bv,
        /*c_mod=*/(short)0, acc, /*reuse_a=*/false, /*reuse_b=*/false);
    ap += 4;
    bp += 4 * 16;
  }

#pragma unroll
  for (int r = 0; r < 8; ++r) {
    const int m = m0 + r + ((lane >> 4) * 8);
    const float val = acc[r];
    if (ncol < 8) ei[m * HEADS + ncol];
    if (ncol < 8) ei[m * HEADS + ncol] = val;
    else          ejt[(size_t)(ncol - 8) * NN + m] = val;
  }
}

// ---------------------------------------------------------------------------
// Kernel 3: streaming attention rows. One block per row i (4096 blocks, 256 thr).
//   norm[h] = sum_j lrelu(ei[h,i]+ej[h,j]) * g[i,j]
//   out[h,i,j] = lrelu(ei[h,i]+ej[h,j]) / norm[h]      (537 MB NT stores)
// ---------------------------------------------------------------------------
__global__ void __launch_bounds__(256)
gat_attn(const float* __restrict__ g, const float* __restrict__ ei,
         const float* __restrict__ ejt, float* __restrict__ out) {
  __shared__ float sg[NN];            // 16 KB: one adjacency row
  __shared__ float red[256 * HEADS];  // 8 KB: norm reduction
  __shared__ float sinv[HEADS];

  const int i = blockIdx.x;
  const int t = threadIdx.x;

  // stage g row into LDS with 128-bit loads
  const v4f* g4  = (const v4f*)(g + (size_t)i * NN);
  v4f*       sg4 = (v4f*)sg;
  for (int c = t; c < NN / 4; c += 256) sg4[c] = __builtin_nontemporal_load(&g4[c]);

  float eih[HEADS];
#pragma unroll
  for (int h = 0; h < HEADS; ++h) eih[h] = ei[i * HEADS + h];
  __syncthreads();

  // pass 1: per-thread partial norms for all 8 heads
  float acc[HEADS];
#pragma unroll
  for (int h = 0; h < HEADS; ++h) acc[h] = 0.0f;
  for (int c = t; c < NN / 4; c += 256) {
    const v4f gv = sg4[c];
#pragma unroll
    for (int h = 0; h < HEADS; ++h) {
      const v4f ej = ((const v4f*)(ejt + (size_t)h * NN))[c];
      const float s0 = lrelu(eih[h] + ej.x);
      const float s1 = lrelu(eih[h] + ej.y);
      const float s2 = lrelu(eih[h] + ej.z);
      const float s3 = lrelu(eih[h] + ej.w);
      acc[h] += s0 * gv.x + s1 * gv.y + s2 * gv.z + s3 * gv.w;
    }
  }
#pragma unroll
  for (int h = 0; h < HEADS; ++h) red[t * HEADS + h] = acc[h];
  __syncthreads();
  for (int s = 128; s > 0; s >>= 1) {
    if (t < s) {
#pragma unroll
      for (int h = 0; h < HEADS; ++h)
        red[t * HEADS + h] += red[(t + s) * HEADS + h];
    }
    __syncthreads();
  }
  if (t < HEADS) sinv[t] = 1.0f / red[t];
  __syncthreads();

  float inv[HEADS];
#pragma unroll
  for (int h = 0; h < HEADS; ++h) inv[h] = sinv[h];

  // pass 2: stream 8 output rows (128 KB) with non-temporal 128-bit stores
  for (int c = t; c < NN / 4; c += 256) {
#pragma unroll
    for (int h = 0; h < HEADS; ++h) {
      const v4f ej = ((const v4f*)(ejt + (size_t)h * NN))[c];
      v4f o;
      o.x = lrelu(eih[h] + ej.x) * inv[h];
      o.y = lrelu(eih[h] + ej.y) * inv[h];
      o.z = lrelu(eih[h] + ej.z) * inv[h];
      o.w = lrelu(eih[h] + ej.w) * inv[h];
      v4f* dst = (v4f*)(out + ((size_t)h * NN + i) * NN);
      __builtin_nontemporal_store(o, &dst[c]);
    }
  }
}

// ---------------------------------------------------------------------------
// d_in order per setup_inputs(): g (B*N*N), x (B*N*F_IN), w (F_IN*F*H), a (2F)
// d_ws layout (floats): WA[0..4095] | EI[4096..36863] | EJT[36864..69631]
//   total 69632 floats = 272 KB scratch.
// ---------------------------------------------------------------------------
extern "C" void kernel_launch(void* const* d_in, const int* in_sizes, int n_in,
                              void* d_out, int out_size, void* d_ws, size_t ws_size,
                              hipStream_t stream) {
  const float* g = (const float*)d_in[0];
  const float* x = (const float*)d_in[1];
  const float* w = (const float*)d_in[2];
  const float* a = (const float*)d_in[3];
  float* out = (float*)d_out;

  float* ws  = (float*)d_ws;
  float* wa  = ws;                 // 256*16
  float* ei  = ws + 4096;          // 4096*8
  float* ejt = ws + 4096 + 32768;  // 8*4096

  gat_wa<<<1, 256, 0, stream>>>(w, a, wa);
  gat_proj<<<NN / 16, 32, 0, stream>>>(x, wa, ei, ejt);
  gat_attn<<<NN, 256, 0, stream>>>(g, ei, ejt, out);
}